// GlowTTS_54941221650787
// MI455X (gfx1250) — compile-verified
//
#include <hip/hip_runtime.h>
#include <hip/hip_bf16.h>

typedef __attribute__((ext_vector_type(2))) float v2f;
typedef __attribute__((ext_vector_type(8))) float v8f;

namespace {
constexpr int B = 32, T = 1200, S = 384, C = 160;
constexpr int S32 = S / 32;  // direction words per (b,t) row
constexpr float kLog2Pi = 1.8378770664093453f;
constexpr float kFMin   = -3.4028234663852886e+38f;  // np.finfo(f32).min
constexpr int LSTR = 20;     // LDS tile row stride (floats): 80B, 16B-aligned rows

#define ASYNC_LDS_B128(ldsaddr, gaddr)                                   \
  asm volatile("global_load_async_to_lds_b128 %0, %1, off"               \
               :: "v"(ldsaddr), "v"(gaddr) : "memory")
#define WAIT_ASYNCCNT_0() asm volatile("s_wait_asynccnt 0x0" ::: "memory")

// ---------------- K1: squared row norms of latent (B*T rows) and mean (B*S rows)
__global__ __launch_bounds__(256)
void k_norms(const float* __restrict__ latent, const float* __restrict__ mean,
             float* __restrict__ lsq, float* __restrict__ msq) {
  __shared__ float red[256];
  int row = blockIdx.x;
  int tid = threadIdx.x;
  const float* src;
  float* dst;
  if (row < B * T) { src = latent + (size_t)row * C; dst = lsq + row; }
  else { row -= B * T; src = mean + (size_t)row * C; dst = msq + row; }
  float v = 0.f;
  if (tid < C) { float x = src[tid]; v = x * x; }
  red[tid] = v; __syncthreads();
  for (int off = 128; off > 0; off >>= 1) {
    if (tid < off) red[tid] += red[tid + off];
    __syncthreads();
  }
  if (tid == 0) *dst = red[0];
}

// ---------------- K2: ll[b,t,s] = attnmask * -0.5*(LOG2PI + ||lat||^2 + ||mean||^2 - 2*dot)
// f32 WMMA GEMM, 64x64 tile per workgroup, 8 waves x two 16x16 subtiles each.
// Tiles staged with async global->LDS B128 DMA, double buffered over K.
__global__ __launch_bounds__(256)
void k_ll_wmma(const float* __restrict__ latent, const float* __restrict__ mean,
               const float* __restrict__ lsq, const float* __restrict__ msq,
               const int* __restrict__ textlen, const int* __restrict__ mellen,
               float* __restrict__ ll) {
  __shared__ float As[2][64 * LSTR];  // [trow][kc] row-major, stride 20
  __shared__ float Bs[2][64 * LSTR];  // [srow][kc] row-major (read transposed)
  __shared__ float Dump[4];           // DMA sink for out-of-range T rows
  const int b   = blockIdx.z;
  const int t0  = blockIdx.x * 64;
  const int s0  = blockIdx.y * 64;
  const int tid = threadIdx.x;
  const int lane = tid & 31;
  const int wave = tid >> 5;
  const int msub  = wave & 3;          // t-subtile 0..3
  const int nbase = (wave >> 2) * 2;   // n-subtiles {0,1} or {2,3}
  const int half = lane >> 4;          // 0 or 1
  const int l15  = lane & 15;
  const int ml = mellen[b], tl = textlen[b];

  // Per-thread DMA assignment: one 16B chunk of each tile.
  const int row = tid >> 2;            // 0..63
  const int q   = (tid & 3) * 4;       // float offset within the 16-float K slice
  const bool tvalid = (t0 + row) < T;

  // Pre-zero out-of-range A rows; DMAs below never touch them (LDS dest -> Dump).
  if (!tvalid) {
    #pragma unroll
    for (int j = 0; j < 4; ++j) {
      As[0][row * LSTR + q + j] = 0.f;
      As[1][row * LSTR + q + j] = 0.f;
    }
  }

  const unsigned ldsDump = (unsigned)(uintptr_t)&Dump[0];
  const unsigned ldsA[2] = {
      tvalid ? (unsigned)(uintptr_t)&As[0][row * LSTR + q] : ldsDump,
      tvalid ? (unsigned)(uintptr_t)&As[1][row * LSTR + q] : ldsDump};
  const unsigned ldsB[2] = {
      (unsigned)(uintptr_t)&Bs[0][row * LSTR + q],
      (unsigned)(uintptr_t)&Bs[1][row * LSTR + q]};
  const float* gA = latent + ((size_t)b * T + (tvalid ? t0 + row : 0)) * C + q;
  const float* gB = mean   + ((size_t)b * S + s0 + row) * C + q;

  auto issue = [&](int buf, int kt) {  // full-EXEC, 2 DMAs per thread per stage
    uint64_t ga = (uint64_t)(uintptr_t)(gA + kt * 16);
    uint64_t gb = (uint64_t)(uintptr_t)(gB + kt * 16);
    ASYNC_LDS_B128(ldsA[buf], ga);
    ASYNC_LDS_B128(ldsB[buf], gb);
  };

  v8f acc0 = {0,0,0,0,0,0,0,0};
  v8f acc1 = {0,0,0,0,0,0,0,0};

  issue(0, 0);
  for (int kt = 0; kt < C / 16; ++kt) {
    const int cur = kt & 1;
    WAIT_ASYNCCNT_0();     // own stage-kt DMAs done (kt+1 not yet issued)
    __syncthreads();       // all waves' DMAs done; also fences prior buf reads
    if (kt + 1 < C / 16) issue(1 - cur, kt + 1);  // overlap next stage with WMMA
    const float* Ac = As[cur];
    const float* Bc = Bs[cur];
    #pragma unroll
    for (int kk = 0; kk < 4; ++kk) {
      const int k0 = kk * 4 + half * 2;  // f32 A 16x4 layout: lanes16-31 hold K=2,3
      v2f a;
      a.x = Ac[(msub * 16 + l15) * LSTR + k0];
      a.y = Ac[(msub * 16 + l15) * LSTR + k0 + 1];
      v2f b0, b1;
      b0.x = Bc[(nbase * 16 + l15) * LSTR + k0];
      b0.y = Bc[(nbase * 16 + l15) * LSTR + k0 + 1];
      b1.x = Bc[((nbase + 1) * 16 + l15) * LSTR + k0];
      b1.y = Bc[((nbase + 1) * 16 + l15) * LSTR + k0 + 1];
      acc0 = __builtin_amdgcn_wmma_f32_16x16x4_f32(false, a, false, b0,
                                                   (short)0, acc0, false, false);
      acc1 = __builtin_amdgcn_wmma_f32_16x16x4_f32(false, a, false, b1,
                                                   (short)0, acc1, false, false);
    }
    // Pin the pipeline stage: keep stage-kt WMMAs here so the stage-(kt+1)
    // DMAs issued above overlap them instead of being waited on immediately.
    __builtin_amdgcn_sched_barrier(0);
  }
  // Epilogue: C/D layout — VGPR r: lanes0-15 M=r, lanes16-31 M=r+8; N = lane&15
  #pragma unroll
  for (int nb = 0; nb < 2; ++nb) {
    v8f acc = nb ? acc1 : acc0;
    int s = s0 + (nbase + nb) * 16 + l15;
    float ms = msq[b * S + s];
    bool smask = (s < tl);
    #pragma unroll
    for (int r = 0; r < 8; ++r) {
      int t = t0 + msub * 16 + r + half * 8;
      if (t < T) {
        float dist = lsq[b * T + t] + ms - 2.f * acc[r];
        float v = (smask && t < ml) ? -0.5f * (kLog2Pi + dist) : 0.f;
        ll[((size_t)b * T + t) * S + s] = v;
      }
    }
  }
}

// ---------------- K3: MAS forward scan; direction packed as ballot bits
__global__ __launch_bounds__(384)
void k_mas_fwd(const float* __restrict__ ll, const int* __restrict__ textlen,
               const int* __restrict__ mellen, unsigned* __restrict__ dir) {
  __shared__ float prob[2][S];
  const int b = blockIdx.x;
  const int s = threadIdx.x;
  const int ml = mellen[b], tl = textlen[b];
  prob[0][s] = 0.f;
  __syncthreads();
  int cur = 0;
  for (int t = 0; t < T; ++t) {
    float llv  = ll[((size_t)b * T + t) * S + s];
    float p    = prob[cur][s];
    float prev = (s == 0) ? -INFINITY : prob[cur][s - 1];
    bool  mm   = (p >= prev);
    float pm   = mm ? p : prev;
    prob[cur ^ 1][s] = (s <= t) ? (pm + llv) : kFMin;
    bool bit = (t < ml && s < tl) ? mm : true;  // where(attnmask, dir, 1)
    unsigned m = __builtin_amdgcn_ballot_w32(bit);
    if ((s & 31) == 0) dir[((size_t)b * T + t) * S32 + (s >> 5)] = m;
    __syncthreads();
    cur ^= 1;
  }
}

// ---------------- K4: serial traceback (descending-j append order) + histogram
__global__ void k_traceback(const unsigned* __restrict__ dir,
                            const int* __restrict__ textlen,
                            const int* __restrict__ mellen,
                            int* __restrict__ sidx, int* __restrict__ cnt) {
  int b = threadIdx.x;
  if (b >= B) return;
  for (int s = 0; s < S; ++s) cnt[b * S + s] = 0;
  int idx = textlen[b] - 1;     // attnmask[:,0].sum(-1)-1 (melmask[:,0]==1 here)
  int ml = mellen[b];
  for (int i = 0; i < T; ++i) { // output row i corresponds to j = T-1-i
    int j = T - 1 - i;
    int out = -1;
    if (idx >= 0) {             // once negative, one_hot rows stay zero
      out = idx;
      unsigned w = dir[((size_t)b * T + j) * S32 + (idx >> 5)];
      int d = (w >> (idx & 31)) & 1;
      idx += d - 1;
    }
    if (i >= ml) out = -1;      // attn *= attnmask (row mask)
    sidx[b * T + i] = out;
    if (out >= 0) cnt[b * S + out] += 1;
  }
}

// ---------------- K5: materialize attn output (one-hot rows)
__global__ __launch_bounds__(256)
void k_attn_write(const int* __restrict__ sidx, float* __restrict__ attn) {
  size_t i = (size_t)blockIdx.x * 256 + threadIdx.x;
  int s = (int)(i % S);
  size_t bt = i / S;
  attn[i] = (s == sidx[bt]) ? 1.f : 0.f;
}

// ---------------- K6: per-(b,t) sum_c (latent - gathered mean)^2
__global__ __launch_bounds__(256)
void k_nll_rows(const float* __restrict__ latent, const float* __restrict__ mean,
                const int* __restrict__ sidx, float* __restrict__ ptsum) {
  __shared__ float red[256];
  int bt = blockIdx.x;
  int b = bt / T;
  int tid = threadIdx.x;
  int sv = sidx[bt];
  float v = 0.f;
  if (tid < C) {
    float a = latent[(size_t)bt * C + tid];
    float m = (sv >= 0) ? mean[((size_t)b * S + sv) * C + tid] : 0.f;
    float d = a - m;
    v = d * d;
  }
  red[tid] = v; __syncthreads();
  for (int off = 128; off > 0; off >>= 1) {
    if (tid < off) red[tid] += red[tid + off];
    __syncthreads();
  }
  if (tid == 0) ptsum[bt] = red[0];
}

// ---------------- K7: per-b nll
__global__ __launch_bounds__(256)
void k_nll_reduce(const float* __restrict__ ptsum, const int* __restrict__ mellen,
                  float* __restrict__ nll_b) {
  __shared__ float red[256];
  int b = blockIdx.x, tid = threadIdx.x;
  float v = 0.f;
  for (int t = tid; t < T; t += 256) v += ptsum[b * T + t];
  red[tid] = v; __syncthreads();
  for (int off = 128; off > 0; off >>= 1) {
    if (tid < off) red[tid] += red[tid + off];
    __syncthreads();
  }
  if (tid == 0) nll_b[b] = 0.5f * (kLog2Pi + red[0]) / ((float)mellen[b] * (float)C);
}

// ---------------- K8: per-b duration loss
__global__ __launch_bounds__(256)
void k_durloss(const float* __restrict__ logdur, const int* __restrict__ cnt,
               const int* __restrict__ textlen, const int* __restrict__ mellen,
               float* __restrict__ dur_b) {
  __shared__ float red[256];
  int b = blockIdx.x, tid = threadIdx.x;
  int tl = textlen[b];
  float v = 0.f;
  for (int s = tid; s < S; s += 256) {
    float gt = (s < tl) ? logf(fmaxf((float)cnt[b * S + s], 1e-5f)) : 0.f;
    float d = logdur[b * S + s] - gt;
    v += d * d;
  }
  red[tid] = v; __syncthreads();
  for (int off = 128; off > 0; off >>= 1) {
    if (tid < off) red[tid] += red[tid + off];
    __syncthreads();
  }
  if (tid == 0) dur_b[b] = red[0] / (float)mellen[b];
}

// ---------------- K9: loss = mean(nll_b) + mean(dur_b)
__global__ void k_final(const float* __restrict__ nll_b,
                        const float* __restrict__ dur_b, float* __restrict__ out) {
  __shared__ float red[32];
  int tid = threadIdx.x;
  red[tid] = nll_b[tid] + dur_b[tid];
  __syncthreads();
  for (int off = 16; off > 0; off >>= 1) {
    if (tid < off) red[tid] += red[tid + off];
    __syncthreads();
  }
  if (tid == 0) out[0] = red[0] / (float)B;
}
}  // namespace

extern "C" void kernel_launch(void* const* d_in, const int* in_sizes, int n_in,
                              void* d_out, int out_size, void* d_ws, size_t ws_size,
                              hipStream_t stream) {
  const float* latent  = (const float*)d_in[0];
  const float* mean    = (const float*)d_in[1];
  const float* logdur  = (const float*)d_in[2];
  const int*   textlen = (const int*)d_in[3];
  const int*   mellen  = (const int*)d_in[4];
  float* out = (float*)d_out;            // [0] = loss, [1..] = attn (B*T*S)

  char* ws = (char*)d_ws;
  size_t o = 0;
  float*    ll    = (float*)(ws + o);    o += (size_t)B * T * S  * 4;  // 56.25 MB
  unsigned* dir   = (unsigned*)(ws + o); o += (size_t)B * T * S32 * 4; // 1.76 MB
  float*    lsq   = (float*)(ws + o);    o += (size_t)B * T * 4;
  float*    msq   = (float*)(ws + o);    o += (size_t)B * S * 4;
  int*      sidx  = (int*)(ws + o);      o += (size_t)B * T * 4;
  int*      cnt   = (int*)(ws + o);      o += (size_t)B * S * 4;
  float*    ptsum = (float*)(ws + o);    o += (size_t)B * T * 4;
  float*    nll_b = (float*)(ws + o);    o += (size_t)B * 4;
  float*    dur_b = (float*)(ws + o);    o += (size_t)B * 4;
  (void)o; (void)ws_size; (void)in_sizes; (void)n_in; (void)out_size;

  k_norms<<<dim3(B * T + B * S), 256, 0, stream>>>(latent, mean, lsq, msq);
  k_ll_wmma<<<dim3((T + 63) / 64, S / 64, B), 256, 0, stream>>>(
      latent, mean, lsq, msq, textlen, mellen, ll);
  k_mas_fwd<<<dim3(B), 384, 0, stream>>>(ll, textlen, mellen, dir);
  k_traceback<<<dim3(1), 32, 0, stream>>>(dir, textlen, mellen, sidx, cnt);
  k_attn_write<<<dim3((B * T * S) / 256), 256, 0, stream>>>(sidx, out + 1);
  k_nll_rows<<<dim3(B * T), 256, 0, stream>>>(latent, mean, sidx, ptsum);
  k_nll_reduce<<<dim3(B), 256, 0, stream>>>(ptsum, mellen, nll_b);
  k_durloss<<<dim3(B), 256, 0, stream>>>(logdur, cnt, textlen, mellen, dur_b);
  k_final<<<dim3(1), 32, 0, stream>>>(nll_b, dur_b, out);
}